// VoxelSetAbstraction_23381801959664
// MI455X (gfx1250) — compile-verified
//
#include <hip/hip_runtime.h>
#include <math.h>

// ---------------- problem constants ----------------
constexpr int Bc = 2, Kc = 2048, Nc = 16384;
constexpr int ROWS = Bc * Kc;          // 4096
constexpr int CBEV = 256, HBEV = 100, WBEV = 176;
constexpr int NS = 16;                 // NSAMPLE
constexpr int FDIM = 304;              // 256 + 16 + 32
constexpr int FOUT = 128;
constexpr float EPSF = 1e-5f;

// ---------------- workspace layout (in floats) ----------------
constexpr size_t OFF_G0     = 0;                                   // grouped scale0: ROWS*16*4
constexpr size_t OFF_G1     = OFF_G0 + (size_t)ROWS * NS * 4;      // grouped scale1
constexpr size_t OFF_E0     = OFF_G1 + (size_t)ROWS * NS * 4;      // empty flags (int) scale0
constexpr size_t OFF_E1     = OFF_E0 + ROWS;
constexpr size_t OFF_ST0_S0 = OFF_E1 + ROWS;                       // layer0 stats s0: sum[16],sq[16]
constexpr size_t OFF_ST0_S1 = OFF_ST0_S0 + 32;
constexpr size_t OFF_ST1_S0 = OFF_ST0_S1 + 32;                     // layer1 stats s0: 16+16
constexpr size_t OFF_ST1_S1 = OFF_ST1_S0 + 32;                     // layer1 stats s1: 32+32
constexpr size_t OFF_FEATS  = OFF_ST1_S1 + 64;                     // ROWS*304
constexpr size_t OFF_Y      = OFF_FEATS + (size_t)ROWS * FDIM;     // ROWS*128
constexpr size_t OFF_STATF  = OFF_Y + (size_t)ROWS * FOUT;         // 128*2

// ================= K1: wave-per-keypoint ball query + grouping + layer0 stats =================
// 32 lanes test 32 points/step; ordered first-16 insertion via ballot + prefix popcount
// (ballot mask is wave-uniform so cnt stays scalar; per-lane slot = cnt + popc(mask & lt)).
__global__ void k1_group(const float* __restrict__ kp, const float* __restrict__ pxyz,
                         const float* __restrict__ pfeat,
                         const float* __restrict__ W0s0, const float* __restrict__ W0s1,
                         float* __restrict__ ws)
{
    constexpr int TILE = 2048;                    // 32 KB point tile
    __shared__ float4 pts[TILE];
    __shared__ int    lists[8][32];               // per wave: 16 (scale0) + 16 (scale1)
    __shared__ float  sW0[2][64];
    __shared__ float  sSum[2][16], sSq[2][16];

    const int tid  = threadIdx.x;
    const int lane = tid & 31;
    const int wv   = tid >> 5;                    // 0..7
    const int b    = blockIdx.y;
    const int k    = blockIdx.x * 8 + wv;         // one keypoint per wave

    for (int i = tid; i < 64; i += 256) { sW0[0][i] = W0s0[i]; sW0[1][i] = W0s1[i]; }
    if (tid < 16) { sSum[0][tid] = 0.f; sSq[0][tid] = 0.f; sSum[1][tid] = 0.f; sSq[1][tid] = 0.f; }

    const size_t kb = ((size_t)b * Kc + k) * 3;
    const float kx = kp[kb + 0], ky = kp[kb + 1], kz = kp[kb + 2];

    int cnt0 = 0, cnt1 = 0;                       // wave-uniform
    int* l0 = lists[wv];
    int* l1 = lists[wv] + 16;
    const unsigned ltmask = (1u << lane) - 1u;

    for (int t0 = 0; t0 < Nc; t0 += TILE) {
        __syncthreads();
        for (int j = tid; j < TILE; j += 256) {
            size_t gj = (size_t)b * Nc + t0 + j;
            float4 p;
            p.x = pxyz[gj * 3 + 0]; p.y = pxyz[gj * 3 + 1]; p.z = pxyz[gj * 3 + 2];
            p.w = pfeat[gj];
            pts[j] = p;
        }
        __syncthreads();
        if ((cnt0 >= NS) && (cnt1 >= NS)) continue;   // wave-uniform skip (still does barriers)
        for (int jj = 0; jj < TILE; jj += 32) {
            float4 p = pts[jj + lane];
            float dx = p.x - kx, dy = p.y - ky, dz = p.z - kz;
            float d2 = dx * dx + dy * dy + dz * dz;
            bool w1 = d2 < 0.64f;                     // radius 0.8
            bool w0 = d2 < 0.16f;                     // radius 0.4
            unsigned m1 = __builtin_amdgcn_ballot_w32(w1);
            unsigned m0 = __builtin_amdgcn_ballot_w32(w0);
            if (w1) {
                int pos = cnt1 + __popc(m1 & ltmask);
                if (pos < NS) l1[pos] = t0 + jj + lane;
            }
            if (w0) {
                int pos = cnt0 + __popc(m0 & ltmask);
                if (pos < NS) l0[pos] = t0 + jj + lane;
            }
            cnt1 += __popc(m1);
            cnt0 += __popc(m0);
            if ((cnt0 >= NS) && (cnt1 >= NS)) break;  // wave-uniform
        }
    }

    // lanes 0-15: scale0 sample s=lane; lanes 16-31: scale1 sample s=lane-16
    const int half = lane >> 4;
    const int s    = lane & 15;
    const int cnt  = half ? cnt1 : cnt0;
    const int* list = half ? l1 : l0;
    const int row  = b * Kc + k;

    float gx = 0.f, gy = 0.f, gz = 0.f, gw = 0.f;
    if (cnt > 0) {
        int gi = (s < cnt) ? list[s] : list[0];       // pcdet pad with first neighbor
        size_t pb = (size_t)b * Nc + gi;
        gx = pxyz[pb * 3 + 0] - kx;
        gy = pxyz[pb * 3 + 1] - ky;
        gz = pxyz[pb * 3 + 2] - kz;
        gw = pfeat[pb];
    }
    if (s == 0) ((int*)(ws + (half ? OFF_E1 : OFF_E0)))[row] = (cnt == 0) ? 1 : 0;
    float4 gv; gv.x = gx; gv.y = gy; gv.z = gz; gv.w = gw;
    *(float4*)(ws + (half ? OFF_G1 : OFF_G0) + (size_t)row * NS * 4 + s * 4) = gv;

    // layer-0 pre-BN stats: reduce over the 16 samples of this half-wave
    const float* W0h = half ? sW0[1] : sW0[0];
    #pragma unroll
    for (int o = 0; o < 16; ++o) {
        float y = gx * W0h[o * 4 + 0] + gy * W0h[o * 4 + 1] + gz * W0h[o * 4 + 2] + gw * W0h[o * 4 + 3];
        float q = y * y;
        #pragma unroll
        for (int m = 1; m < 16; m <<= 1) {
            y += __shfl_xor(y, m, 16);
            q += __shfl_xor(q, m, 16);
        }
        if (s == 0) { atomicAdd(&sSum[half][o], y); atomicAdd(&sSq[half][o], q); }
    }
    __syncthreads();
    if (tid < 16) {
        atomicAdd(&ws[OFF_ST0_S0 + tid],      sSum[0][tid]);
        atomicAdd(&ws[OFF_ST0_S0 + 16 + tid], sSq[0][tid]);
        atomicAdd(&ws[OFF_ST0_S1 + tid],      sSum[1][tid]);
        atomicAdd(&ws[OFF_ST0_S1 + 16 + tid], sSq[1][tid]);
    }
}

// ================= K2: layer-1 stats, thread per (keypoint, sample) =================
template<int COUT>
__global__ void k2_stats(const float* __restrict__ grouped, const float* __restrict__ W0,
                         const float* __restrict__ g0v, const float* __restrict__ b0v,
                         const float* __restrict__ W1,
                         const float* __restrict__ stat0, float* __restrict__ stat1)
{
    __shared__ float sW0[64];
    __shared__ float sW1[COUT * 16];
    __shared__ float sA0[16], sS0[16];
    __shared__ float sSum[COUT], sSq[COUT];
    const int tid = threadIdx.x;
    const int lane = tid & 31;
    for (int i = tid; i < 64; i += 256) sW0[i] = W0[i];
    for (int i = tid; i < COUT * 16; i += 256) sW1[i] = W1[i];
    constexpr float invM = 1.0f / (float)(ROWS * NS);
    if (tid < 16) {
        float mean = stat0[tid] * invM;
        float var  = stat0[16 + tid] * invM - mean * mean;
        float sc   = g0v[tid] * rsqrtf(var + EPSF);
        sA0[tid] = sc; sS0[tid] = b0v[tid] - mean * sc;
    }
    if (tid < COUT) { sSum[tid] = 0.f; sSq[tid] = 0.f; }
    __syncthreads();

    const int gidx = blockIdx.x * 256 + tid;       // (r,s) pair, 0..65535
    const float* g = grouped + (size_t)gidx * 4;
    const float gx = g[0], gy = g[1], gz = g[2], gw = g[3];
    float h[16];
    #pragma unroll
    for (int c = 0; c < 16; ++c) {
        float y = gx * sW0[c * 4 + 0] + gy * sW0[c * 4 + 1] + gz * sW0[c * 4 + 2] + gw * sW0[c * 4 + 3];
        h[c] = fmaxf(y * sA0[c] + sS0[c], 0.f);
    }
    #pragma unroll
    for (int o = 0; o < COUT; ++o) {
        float y = 0.f;
        #pragma unroll
        for (int i = 0; i < 16; ++i) y += h[i] * sW1[o * 16 + i];
        float q = y * y;
        #pragma unroll
        for (int m = 1; m < 32; m <<= 1) { y += __shfl_xor(y, m, 32); q += __shfl_xor(q, m, 32); }
        if (lane == 0) { atomicAdd(&sSum[o], y); atomicAdd(&sSq[o], q); }
    }
    __syncthreads();
    if (tid < COUT) { atomicAdd(&stat1[tid], sSum[tid]); atomicAdd(&stat1[COUT + tid], sSq[tid]); }
}

// ================= K3: MLP apply + shuffle max-pool, thread per (keypoint, sample) =================
template<int COUT>
__global__ void k3_pool(const float* __restrict__ grouped, const int* __restrict__ emptyf,
                        const float* __restrict__ W0, const float* __restrict__ g0v, const float* __restrict__ b0v,
                        const float* __restrict__ W1, const float* __restrict__ g1v, const float* __restrict__ b1v,
                        const float* __restrict__ stat0, const float* __restrict__ stat1,
                        float* __restrict__ feats, int coloff)
{
    __shared__ float sW0[64];
    __shared__ float sW1[COUT * 16];
    __shared__ float sA0[16], sS0[16];
    __shared__ float sA1[COUT], sS1[COUT];
    const int tid = threadIdx.x;
    for (int i = tid; i < 64; i += 256) sW0[i] = W0[i];
    for (int i = tid; i < COUT * 16; i += 256) sW1[i] = W1[i];
    constexpr float invM = 1.0f / (float)(ROWS * NS);
    if (tid < 16) {
        float mean = stat0[tid] * invM;
        float var  = stat0[16 + tid] * invM - mean * mean;
        float sc   = g0v[tid] * rsqrtf(var + EPSF);
        sA0[tid] = sc; sS0[tid] = b0v[tid] - mean * sc;
    }
    if (tid < COUT) {
        float mean = stat1[tid] * invM;
        float var  = stat1[COUT + tid] * invM - mean * mean;
        float sc   = g1v[tid] * rsqrtf(var + EPSF);
        sA1[tid] = sc; sS1[tid] = b1v[tid] - mean * sc;
    }
    __syncthreads();

    const int gidx = blockIdx.x * 256 + tid;      // (r,s)
    const int r = gidx >> 4;
    const int s = gidx & 15;                      // sample id == lane position within width-16 group
    const float* g = grouped + (size_t)gidx * 4;
    const float gx = g[0], gy = g[1], gz = g[2], gw = g[3];
    float h[16];
    #pragma unroll
    for (int c = 0; c < 16; ++c) {
        float y = gx * sW0[c * 4 + 0] + gy * sW0[c * 4 + 1] + gz * sW0[c * 4 + 2] + gw * sW0[c * 4 + 3];
        h[c] = fmaxf(y * sA0[c] + sS0[c], 0.f);
    }
    float keep0 = 0.f, keep1 = 0.f;
    #pragma unroll
    for (int o = 0; o < COUT; ++o) {
        float y = 0.f;
        #pragma unroll
        for (int i = 0; i < 16; ++i) y += h[i] * sW1[o * 16 + i];
        float z = fmaxf(y * sA1[o] + sS1[o], 0.f);
        #pragma unroll
        for (int m = 1; m < 16; m <<= 1) z = fmaxf(z, __shfl_xor(z, m, 16));  // max over 16 samples
        if (o == s)      keep0 = z;
        if (o == s + 16) keep1 = z;
    }
    const int emp = emptyf[r];
    feats[(size_t)r * FDIM + coloff + s] = emp ? 0.f : keep0;
    if (COUT == 32)
        feats[(size_t)r * FDIM + coloff + 16 + s] = emp ? 0.f : keep1;
}

// ================= K4: BEV bilinear interpolation -> feats[:,0:256] =================
__global__ void k4_bev(const float* __restrict__ kp, const float* __restrict__ sf,
                       const int* __restrict__ stridep, float* __restrict__ feats)
{
    const int r = blockIdx.x;          // 0..ROWS-1
    const int b = r >> 11;             // /Kc
    const int c = threadIdx.x;         // 0..255
    const float stride = (float)(*stridep);
    const float x  = (kp[(size_t)r * 3 + 0] - (-70.4f)) / 0.1f / stride;
    const float yy = (kp[(size_t)r * 3 + 1] - (-40.0f)) / 0.1f / stride;
    int xf = (int)floorf(x), yf = (int)floorf(yy);
    int x0 = min(max(xf, 0), WBEV - 1);
    int x1 = min(max(xf + 1, 0), WBEV - 1);
    int y0 = min(max(yf, 0), HBEV - 1);
    int y1 = min(max(yf + 1, 0), HBEV - 1);
    float x0f = (float)x0, x1f = (float)x1, y0f = (float)y0, y1f = (float)y1;
    float wa = (x1f - x) * (y1f - yy);
    float wb = (x1f - x) * (yy - y0f);
    float wc = (x - x0f) * (y1f - yy);
    float wd = (x - x0f) * (yy - y0f);
    const float* base = sf + (((size_t)b * CBEV + c) * HBEV) * WBEV;
    float Ia = base[y0 * WBEV + x0], Ib = base[y1 * WBEV + x0];
    float Ic = base[y0 * WBEV + x1], Id = base[y1 * WBEV + x1];
    feats[(size_t)r * FDIM + c] = Ia * wa + Ib * wb + Ic * wc + Id * wd;
}

// ================= K5: fused GEMM (4096x304)@(304x128)^T, f32 WMMA 16x16x4 =================
typedef float v2f  __attribute__((ext_vector_type(2)));
typedef float v8f  __attribute__((ext_vector_type(8)));

__global__ void k5_gemm(const float* __restrict__ feats, const float* __restrict__ Wf,
                        float* __restrict__ y)
{
    const int lane = threadIdx.x & 31;
    const int wave = threadIdx.x >> 5;
    const int tile = blockIdx.x * (blockDim.x >> 5) + wave;   // 0..2047
    const int m0 = (tile >> 3) * 16;                          // 256 row-tiles
    const int n0 = (tile & 7) * 16;                           // 8 col-tiles
    const int lm = lane & 15;
    const int kh = (lane >> 4) << 1;                          // 0 (lanes 0-15) or 2 (lanes 16-31)
    const float* arow = feats + (size_t)(m0 + lm) * FDIM;
    const float* brow = Wf    + (size_t)(n0 + lm) * FDIM;
    v8f c = {0.f, 0.f, 0.f, 0.f, 0.f, 0.f, 0.f, 0.f};
    for (int k = 0; k < FDIM; k += 4) {                       // 76 chained WMMAs
        v2f a, b;
        a.x = arow[k + kh];     a.y = arow[k + kh + 1];
        b.x = brow[k + kh];     b.y = brow[k + kh + 1];
        c = __builtin_amdgcn_wmma_f32_16x16x4_f32(false, a, false, b, (short)0, c, false, false);
    }
    const int rbase = m0 + ((lane >> 4) ? 8 : 0);
    #pragma unroll
    for (int i = 0; i < 8; ++i)
        y[(size_t)(rbase + i) * FOUT + n0 + lm] = c[i];
}

// ================= K6: per-column sum / sumsq over 4096 rows =================
__global__ void k6_colstats(const float* __restrict__ y, float* __restrict__ statf)
{
    __shared__ float ssum[256], ssq[256];
    const int n = blockIdx.x, tid = threadIdx.x;
    float s = 0.f, q = 0.f;
    for (int m = tid; m < ROWS; m += 256) {
        float v = y[(size_t)m * FOUT + n];
        s += v; q += v * v;
    }
    ssum[tid] = s; ssq[tid] = q;
    __syncthreads();
    for (int off = 128; off > 0; off >>= 1) {
        if (tid < off) { ssum[tid] += ssum[tid + off]; ssq[tid] += ssq[tid + off]; }
        __syncthreads();
    }
    if (tid == 0) { statf[2 * n] = ssum[0]; statf[2 * n + 1] = ssq[0]; }
}

// ================= K7: batchnorm + relu -> output =================
__global__ void k7_bnrelu(const float* __restrict__ y, const float* __restrict__ statf,
                          const float* __restrict__ gf, const float* __restrict__ bfv,
                          float* __restrict__ out)
{
    const int i = blockIdx.x * 256 + threadIdx.x;   // ROWS*FOUT elements
    const int n = i & (FOUT - 1);
    constexpr float invM = 1.0f / (float)ROWS;
    float mean = statf[2 * n] * invM;
    float var  = statf[2 * n + 1] * invM - mean * mean;
    float a = gf[n] * rsqrtf(var + EPSF);
    float v = (y[i] - mean) * a + bfv[n];
    out[i] = fmaxf(v, 0.f);
}

// ================= host launch =================
extern "C" void kernel_launch(void* const* d_in, const int* in_sizes, int n_in,
                              void* d_out, int out_size, void* d_ws, size_t ws_size,
                              hipStream_t stream) {
    (void)in_sizes; (void)n_in; (void)out_size; (void)ws_size;
    const float* kp    = (const float*)d_in[0];
    const float* pxyz  = (const float*)d_in[1];
    const float* pfeat = (const float*)d_in[2];
    const float* sf    = (const float*)d_in[3];
    const int*   strd  = (const int*)  d_in[4];
    const float* W0s0  = (const float*)d_in[5];
    const float* g0s0  = (const float*)d_in[6];
    const float* b0s0  = (const float*)d_in[7];
    const float* W1s0  = (const float*)d_in[8];
    const float* g1s0  = (const float*)d_in[9];
    const float* b1s0  = (const float*)d_in[10];
    const float* W0s1  = (const float*)d_in[11];
    const float* g0s1  = (const float*)d_in[12];
    const float* b0s1  = (const float*)d_in[13];
    const float* W1s1  = (const float*)d_in[14];
    const float* g1s1  = (const float*)d_in[15];
    const float* b1s1  = (const float*)d_in[16];
    const float* Wf    = (const float*)d_in[17];
    const float* gf    = (const float*)d_in[18];
    const float* bfv   = (const float*)d_in[19];
    float* ws = (float*)d_ws;

    // zero the atomic stats accumulators (160 floats)
    hipMemsetAsync(ws + OFF_ST0_S0, 0, (OFF_FEATS - OFF_ST0_S0) * sizeof(float), stream);

    // 512 blocks x 8 waves = 4096 waves, one keypoint per wave
    k1_group<<<dim3(Kc / 8, Bc), 256, 0, stream>>>(kp, pxyz, pfeat, W0s0, W0s1, ws);

    // thread per (keypoint, sample): 65536 threads = 256 blocks
    k2_stats<16><<<(ROWS * NS) / 256, 256, 0, stream>>>(ws + OFF_G0, W0s0, g0s0, b0s0, W1s0,
                                                        ws + OFF_ST0_S0, ws + OFF_ST1_S0);
    k2_stats<32><<<(ROWS * NS) / 256, 256, 0, stream>>>(ws + OFF_G1, W0s1, g0s1, b0s1, W1s1,
                                                        ws + OFF_ST0_S1, ws + OFF_ST1_S1);

    k3_pool<16><<<(ROWS * NS) / 256, 256, 0, stream>>>(ws + OFF_G0, (const int*)(ws + OFF_E0),
                                                       W0s0, g0s0, b0s0, W1s0, g1s0, b1s0,
                                                       ws + OFF_ST0_S0, ws + OFF_ST1_S0,
                                                       ws + OFF_FEATS, CBEV);
    k3_pool<32><<<(ROWS * NS) / 256, 256, 0, stream>>>(ws + OFF_G1, (const int*)(ws + OFF_E1),
                                                       W0s1, g0s1, b0s1, W1s1, g1s1, b1s1,
                                                       ws + OFF_ST0_S1, ws + OFF_ST1_S1,
                                                       ws + OFF_FEATS, CBEV + 16);

    k4_bev<<<ROWS, CBEV, 0, stream>>>(kp, sf, strd, ws + OFF_FEATS);

    // 2048 output tiles, 4 waves (128 threads) per block
    k5_gemm<<<512, 128, 0, stream>>>(ws + OFF_FEATS, Wf, ws + OFF_Y);

    k6_colstats<<<FOUT, 256, 0, stream>>>(ws + OFF_Y, ws + OFF_STATF);

    k7_bnrelu<<<(ROWS * FOUT) / 256, 256, 0, stream>>>(ws + OFF_Y, ws + OFF_STATF,
                                                       gf, bfv, (float*)d_out);
}